// MultiHeadAttention_42631845380826
// MI455X (gfx1250) — compile-verified
//
#include <hip/hip_runtime.h>
#include <math.h>

typedef float v2f __attribute__((ext_vector_type(2)));
typedef float v8f __attribute__((ext_vector_type(8)));

#define NB    8
#define T_SEQ 2048
#define DM    512
#define NH    8
#define HD    64
#define QKV_ELEMS (NB * NH * T_SEQ * HD)   /* 8,388,608 floats per tensor */

static __device__ __forceinline__ v8f wmma_f32(v2f a, v2f b, v8f c) {
  // D = A(16x4) * B(4x16) + C(16x16), full fp32
  return __builtin_amdgcn_wmma_f32_16x16x4_f32(false, a, false, b, (short)0, c, false, false);
}

// ---------------------------------------------------------------------------
// Kernel 1: QKV projection. qkv[r, e] = sum_d x[r,d] * Wqkv[e,d]
// One wave per 16x16 tile. Scatter into Q/K/V with [n,h,t,d] layout.
// ---------------------------------------------------------------------------
__global__ __launch_bounds__(256) void qkv_kernel(const float* __restrict__ x,
                                                  const float* __restrict__ wqkv,
                                                  float* __restrict__ q,
                                                  float* __restrict__ k,
                                                  float* __restrict__ v) {
  const int wave = threadIdx.x >> 5;
  const int lane = threadIdx.x & 31;
  const int l = lane & 15;
  const int c = lane >> 4;
  const int row0 = blockIdx.x * 16;
  const int col0 = blockIdx.y * 128 + wave * 16;

  const float* arow = x    + (size_t)(row0 + l) * DM;
  const float* brow = wqkv + (size_t)(col0 + l) * DM;

  v8f acc = {};
  #pragma unroll 4
  for (int kb = 0; kb < DM; kb += 4) {
    v2f a = *(const v2f*)(arow + kb + 2 * c);
    v2f b = *(const v2f*)(brow + kb + 2 * c);
    acc = wmma_f32(a, b, acc);
  }

  const int e     = col0 + l;
  const int which = e >> 9;       // 0=q, 1=k, 2=v
  const int rem   = e & 511;
  const int h     = rem >> 6;
  const int d     = rem & 63;
  float* dst = (which == 0) ? q : (which == 1) ? k : v;
  #pragma unroll
  for (int r = 0; r < 8; ++r) {
    const int row = row0 + r + 8 * c;
    const int n = row >> 11;
    const int t = row & (T_SEQ - 1);
    dst[(size_t)(((n * NH + h) * T_SEQ) + t) * HD + d] = acc[r];
  }
}

// ---------------------------------------------------------------------------
// Kernel 2: RoPE applied in place to Q and K ([n,h,t,d] layout).
// Thread handles one (n,h,t,j) pair, j in [0,32), rotating (d=j, d=j+32).
// ---------------------------------------------------------------------------
__global__ __launch_bounds__(256) void rope_kernel(float* __restrict__ q,
                                                   float* __restrict__ k) {
  const int idx  = blockIdx.x * blockDim.x + threadIdx.x;
  const int half = NB * NH * T_SEQ * 32;   // 4,194,304
  float* buf = (idx < half) ? q : k;
  const int id = (idx < half) ? idx : idx - half;
  const int j  = id & 31;
  const int t  = (id >> 5) & (T_SEQ - 1);
  const int nh = id >> 16;                 // n*NH + h
  float* row = buf + (size_t)(nh * T_SEQ + t) * HD;

  const float inv = powf(10000.0f, -(float)j * (1.0f / 32.0f));
  const float ang = (float)t * inv;
  float sn, cs;
  sincosf(ang, &sn, &cs);
  const float a = row[j];
  const float b = row[j + 32];
  row[j]      = a * cs - b * sn;
  row[j + 32] = b * cs + a * sn;
}

// ---------------------------------------------------------------------------
// Kernel 3: banded flash attention. One wave per (n, h, 16-row query tile).
// S = Q K^T (16 wmma), online softmax, P V (16 wmma) via LDS transpose of P.
// Output written to att in [n, t, h*64+d] layout (rows ready for out-proj).
// ---------------------------------------------------------------------------
__global__ __launch_bounds__(128) void attn_kernel(const float* __restrict__ Q,
                                                   const float* __restrict__ K,
                                                   const float* __restrict__ V,
                                                   float* __restrict__ att) {
  __shared__ float pshm[4][256];

  const int wave = threadIdx.x >> 5;
  const int lane = threadIdx.x & 31;
  const int l = lane & 15;
  const int c = lane >> 4;

  const int tile = blockIdx.x * 4 + wave;   // [0, 8192)
  const int it = tile & 127;
  const int h  = (tile >> 7) & (NH - 1);
  const int n  = tile >> 10;
  const int i0 = it * 16;

  const size_t base = (size_t)(n * NH + h) * T_SEQ * HD;
  const float* Qb = Q + base;
  const float* Kb = K + base;
  const float* Vb = V + base;

  // Q fragments for K-dim = 64 (16 steps of 4), scale 1/sqrt(64) folded in.
  v2f qa[16];
  #pragma unroll
  for (int kk = 0; kk < 16; ++kk) {
    v2f t2 = *(const v2f*)(Qb + (size_t)(i0 + l) * HD + 4 * kk + 2 * c);
    t2[0] *= 0.125f;
    t2[1] *= 0.125f;
    qa[kk] = t2;
  }

  float m[8], lsum[8];
  v8f o[4];
  #pragma unroll
  for (int r = 0; r < 8; ++r) { m[r] = -1e30f; lsum[r] = 0.0f; }
  #pragma unroll
  for (int nn = 0; nn < 4; ++nn) o[nn] = (v8f){};

  int jlo = i0 - 127; if (jlo < 0) jlo = 0; jlo &= ~15;
  int jhi = i0 + 15 + 128; if (jhi > T_SEQ - 1) jhi = T_SEQ - 1;

  for (int j0 = jlo; j0 <= jhi; j0 += 16) {
    // ---- S = Q * K^T ----
    v8f s = {};
    #pragma unroll
    for (int kk = 0; kk < 16; ++kk) {
      v2f kb = *(const v2f*)(Kb + (size_t)(j0 + l) * HD + 4 * kk + 2 * c);
      s = wmma_f32(qa[kk], kb, s);
    }

    // ---- mask + online softmax (rows live in 16-lane halves) ----
    float p[8], alpha[8];
    #pragma unroll
    for (int r = 0; r < 8; ++r) {
      const int i = i0 + r + 8 * c;
      const int j = j0 + l;
      float sv = s[r];
      const int dj = j - i;
      if (dj < -127 || dj > 128) sv = -1e30f;

      float rm = sv;
      #pragma unroll
      for (int off = 1; off < 16; off <<= 1)
        rm = fmaxf(rm, __shfl_xor(rm, off, 32));

      const float mn = fmaxf(m[r], rm);
      alpha[r] = __expf(m[r] - mn);
      const float pv = __expf(sv - mn);
      m[r] = mn;

      float rs = pv;
      #pragma unroll
      for (int off = 1; off < 16; off <<= 1)
        rs += __shfl_xor(rs, off, 32);

      lsum[r] = lsum[r] * alpha[r] + rs;
      p[r] = pv;
    }

    // ---- transpose P (D-layout -> A-layout) through LDS ----
    #pragma unroll
    for (int r = 0; r < 8; ++r)
      pshm[wave][(r + 8 * c) * 16 + l] = p[r];
    asm volatile("s_wait_dscnt 0" ::: "memory");
    v2f pa[4];
    #pragma unroll
    for (int kk = 0; kk < 4; ++kk) {
      pa[kk][0] = pshm[wave][l * 16 + 4 * kk + 2 * c];
      pa[kk][1] = pshm[wave][l * 16 + 4 * kk + 2 * c + 1];
    }

    // ---- rescale accumulators, then O += P * V ----
    #pragma unroll
    for (int nn = 0; nn < 4; ++nn)
      #pragma unroll
      for (int r = 0; r < 8; ++r)
        o[nn][r] *= alpha[r];

    #pragma unroll
    for (int nn = 0; nn < 4; ++nn) {
      #pragma unroll
      for (int kk = 0; kk < 4; ++kk) {
        v2f vb;
        vb[0] = Vb[(size_t)(j0 + 4 * kk + 2 * c)     * HD + nn * 16 + l];
        vb[1] = Vb[(size_t)(j0 + 4 * kk + 2 * c + 1) * HD + nn * 16 + l];
        o[nn] = wmma_f32(pa[kk], vb, o[nn]);
      }
    }
  }

  // ---- normalize and write att[n, t, h*64 + d] ----
  #pragma unroll
  for (int nn = 0; nn < 4; ++nn) {
    #pragma unroll
    for (int r = 0; r < 8; ++r) {
      const int t = i0 + r + 8 * c;
      att[((size_t)n * T_SEQ + t) * DM + h * HD + nn * 16 + l] = o[nn][r] / lsum[r];
    }
  }
}

// ---------------------------------------------------------------------------
// Kernel 4: output projection. out[r,e] = sum_d att[r,d]*Wout[e,d] + bout[e]
// ---------------------------------------------------------------------------
__global__ __launch_bounds__(256) void proj_kernel(const float* __restrict__ att,
                                                   const float* __restrict__ wout,
                                                   const float* __restrict__ bout,
                                                   float* __restrict__ out) {
  const int wave = threadIdx.x >> 5;
  const int lane = threadIdx.x & 31;
  const int l = lane & 15;
  const int c = lane >> 4;
  const int row0 = blockIdx.x * 16;
  const int col0 = blockIdx.y * 128 + wave * 16;

  const float* arow = att  + (size_t)(row0 + l) * DM;
  const float* brow = wout + (size_t)(col0 + l) * DM;

  v8f acc = {};
  #pragma unroll 4
  for (int kb = 0; kb < DM; kb += 4) {
    v2f a = *(const v2f*)(arow + kb + 2 * c);
    v2f b = *(const v2f*)(brow + kb + 2 * c);
    acc = wmma_f32(a, b, acc);
  }

  const int e = col0 + l;
  const float bb = bout[e];
  #pragma unroll
  for (int r = 0; r < 8; ++r) {
    const int row = row0 + r + 8 * c;
    out[(size_t)row * DM + e] = acc[r] + bb;
  }
}

// ---------------------------------------------------------------------------
extern "C" void kernel_launch(void* const* d_in, const int* in_sizes, int n_in,
                              void* d_out, int out_size, void* d_ws, size_t ws_size,
                              hipStream_t stream) {
  const float* x    = (const float*)d_in[0];
  const float* wqkv = (const float*)d_in[1];
  const float* wout = (const float*)d_in[2];
  const float* bout = (const float*)d_in[3];
  float* out = (float*)d_out;

  float* q   = (float*)d_ws;
  float* k   = q + QKV_ELEMS;
  float* v   = k + QKV_ELEMS;
  float* att = v + QKV_ELEMS;

  // 1) QKV projection: rows = N*T = 16384, cols = 1536
  qkv_kernel<<<dim3(16384 / 16, 1536 / 128), 256, 0, stream>>>(x, wqkv, q, k, v);

  // 2) RoPE on Q and K: 2 * (8*8*2048*32) threads
  rope_kernel<<<(2 * NB * NH * T_SEQ * 32) / 256, 256, 0, stream>>>(q, k);

  // 3) Banded flash attention: 8*8*128 = 8192 wave-tiles, 4 waves per block
  attn_kernel<<<8192 / 4, 128, 0, stream>>>(q, k, v, att);

  // 4) Output projection: rows = 16384, cols = 512
  proj_kernel<<<dim3(16384 / 16, 512 / 128), 256, 0, stream>>>(att, wout, bout, out);
}